// Llama3Attention_68624987455612
// MI455X (gfx1250) — compile-verified
//
#include <hip/hip_runtime.h>
#include <math.h>

typedef __attribute__((ext_vector_type(2))) float v2f;
typedef __attribute__((ext_vector_type(8))) float v8f;

#define HIDDEN       4096
#define NUM_HEADS    32
#define NUM_KV_HEADS 8
#define HEAD_DIM     128
#define BATCH        8
#define PAST_LEN     4096
#define QKV_ROWS     (NUM_HEADS * HEAD_DIM + 2 * NUM_KV_HEADS * HEAD_DIM) /* 6144 */
#define ROPE_THETA   500000.0f

// ---------------------------------------------------------------------------
// K-split WMMA GEMV:  part[s][m][n] = sum_{k in chunk s} X[m][k] * W[n][k]
// m in [0,8) padded to 16; one wave = one 16-col tile; blockIdx.y = K-chunk,
// so the loop bounds are wave-uniform scalars (keeps the loop unrollable and
// free of EXEC-mask structurization).
// V_WMMA_F32_16X16X4_F32: A 16x4 f32 (2 VGPR), B 4x16 f32 (2 VGPR), C 16x16 f32.
// A layout: lane L holds M=L%16, K = 2*(L/16) + {0,1}  -> contiguous b64 load.
// B layout (symmetric): lane L holds N=L%16, K = 2*(L/16) + {0,1} -> b64 load.
// ---------------------------------------------------------------------------
__global__ __launch_bounds__(256) void gemv16_wmma_split(
    const float* __restrict__ X,    // [8][K]
    const float* __restrict__ W0, int rows0,
    const float* __restrict__ W1, int rows1,
    const float* __restrict__ W2,
    float* __restrict__ part,       // [split][8][Ntot]
    int K, int Ntot, int split)
{
  const int lane = threadIdx.x & 31;
  const int tile = blockIdx.x * 8 + (threadIdx.x >> 5);
  const int sidx = blockIdx.y;               // scalar K-chunk index
  const int n0   = tile * 16;
  if (n0 >= Ntot) return;

  const float* W; int nrel;
  if (n0 < rows0)              { W = W0; nrel = n0; }
  else if (n0 < rows0 + rows1) { W = W1; nrel = n0 - rows0; }
  else                         { W = W2; nrel = n0 - rows0 - rows1; }

  const int   Kc     = K / split;            // scalar
  const int   k0     = sidx * Kc;            // scalar
  const int   half   = lane >> 4;            // 0/1 -> K offset 0/2
  const int   l16    = lane & 15;
  const float mscale = (l16 < BATCH) ? 1.0f : 0.0f;   // zero-pad M rows 8..15
  const int   mrow   = (l16 < BATCH) ? l16 : 0;

  const float* xp = X + (size_t)mrow * K + k0 + 2 * half;
  const float* wp = W + (size_t)(nrel + l16) * K + k0 + 2 * half;

  v8f c = {0.f, 0.f, 0.f, 0.f, 0.f, 0.f, 0.f, 0.f};

#pragma unroll 4
  for (int k = 0; k < Kc; k += 4) {          // uniform scalar trip count
    v2f a = *(const v2f*)(xp + k);
    a = a * mscale;
    v2f b = *(const v2f*)(wp + k);
    c = __builtin_amdgcn_wmma_f32_16x16x4_f32(false, a, false, b,
                                              (short)0, c, false, false);
  }

  // C layout: VGPR r, lanes 0-15 -> C[M=r][N=lane]; M=0..7 are real batches.
  if (half == 0) {
    float* y = part + (size_t)sidx * BATCH * Ntot + n0 + l16;
#pragma unroll
    for (int r = 0; r < BATCH; ++r)
      y[(size_t)r * Ntot] = c[r];
  }
}

// Fixed-order (deterministic) reduction of K-split partials.
__global__ void reduce_split(const float* __restrict__ part,
                             float* __restrict__ Y,
                             int Ntot, int split, int ldy)
{
  int idx = blockIdx.x * blockDim.x + threadIdx.x;
  if (idx >= BATCH * Ntot) return;
  int bb = idx / Ntot, n = idx - bb * Ntot;
  float s = 0.f;
  for (int i = 0; i < split; ++i)
    s += part[((size_t)i * BATCH + bb) * Ntot + n];
  Y[(size_t)bb * ldy + n] = s;
}

// ---------------------------------------------------------------------------
// RoPE at position PAST_LEN on q rows [0,4096) and new-k rows [4096,5120).
// ---------------------------------------------------------------------------
__global__ void rope_kernel(float* __restrict__ qkv)
{
  const int total = BATCH * (NUM_HEADS + NUM_KV_HEADS) * (HEAD_DIM / 2);
  int idx = blockIdx.x * blockDim.x + threadIdx.x;
  if (idx >= total) return;

  int i = idx & 63;
  int h = (idx >> 6) % (NUM_HEADS + NUM_KV_HEADS);
  int b = idx / ((NUM_HEADS + NUM_KV_HEADS) * 64);
  float* base = qkv + (size_t)b * QKV_ROWS + h * HEAD_DIM;

  float inv_freq = powf(ROPE_THETA, -(float)i / 64.0f);
  float ang = (float)PAST_LEN * inv_freq;
  float s, cth;
  sincosf(ang, &s, &cth);

  float x0 = base[i], x1 = base[i + 64];
  base[i]      = x0 * cth - x1 * s;
  base[i + 64] = x1 * cth + x0 * s;
}

// ---------------------------------------------------------------------------
// wave32 reductions
// ---------------------------------------------------------------------------
__device__ __forceinline__ float wred_max(float v) {
#pragma unroll
  for (int off = 16; off > 0; off >>= 1) v = fmaxf(v, __shfl_xor(v, off, 32));
  return v;
}
__device__ __forceinline__ float wred_sum(float v) {
#pragma unroll
  for (int off = 16; off > 0; off >>= 1) v += __shfl_xor(v, off, 32);
  return v;
}

// ---------------------------------------------------------------------------
// Decode attention, 2 q-heads per block (block = (b, kv-head, pair)) so the
// K/V streams are read once for two GQA heads. KV length = PAST_LEN + 1; the
// new token's k/v live in the qkv workspace (inputs never mutated).
// ---------------------------------------------------------------------------
__global__ __launch_bounds__(256) void attn_kernel(
    const float* __restrict__ qkv,
    const float* __restrict__ past_k,  // [8][8][4096][128]
    const float* __restrict__ past_v,
    float* __restrict__ attn_out)      // [8][4096] (h*128+d)
{
  const int b    = blockIdx.x;        // 8
  const int kvh  = blockIdx.y;        // 8
  const int pr   = blockIdx.z;        // 2
  const int h0   = kvh * 4 + pr * 2;  // this block: heads h0, h0+1
  const int tid  = threadIdx.x;
  const int lane = tid & 31;
  const int wid  = tid >> 5;

  __shared__ float q_s[2][HEAD_DIM];
  __shared__ float probs[2][PAST_LEN + 1];
  __shared__ float red0[8], red1[8];
  __shared__ float gsh[4];            // gmax0, gmax1, inv0, inv1
  __shared__ float pv[8][2][HEAD_DIM];

  const float* qbase = qkv + (size_t)b * QKV_ROWS + h0 * HEAD_DIM;
  if (tid < 2 * HEAD_DIM)
    q_s[tid >> 7][tid & 127] = qbase[tid] * 0.08838834764831845f; // 1/sqrt(128)
  __syncthreads();

  // ---- scores: stream K once, score both heads ----
  const float* Kc = past_k + ((size_t)b * NUM_KV_HEADS + kvh) * (size_t)PAST_LEN * HEAD_DIM;
  float lmax0 = -3.4e38f, lmax1 = -3.4e38f;
  for (int j = tid; j < PAST_LEN; j += 256) {
    const float* kr = Kc + (size_t)j * HEAD_DIM;
    float a0 = 0.f, a1 = 0.f;
#pragma unroll
    for (int d = 0; d < HEAD_DIM; d += 4) {
      float4 kk = *(const float4*)(kr + d);
      a0 += kk.x * q_s[0][d] + kk.y * q_s[0][d + 1] + kk.z * q_s[0][d + 2] + kk.w * q_s[0][d + 3];
      a1 += kk.x * q_s[1][d] + kk.y * q_s[1][d + 1] + kk.z * q_s[1][d + 2] + kk.w * q_s[1][d + 3];
    }
    probs[0][j] = a0; probs[1][j] = a1;
    lmax0 = fmaxf(lmax0, a0); lmax1 = fmaxf(lmax1, a1);
  }
  if (tid == 0) {  // new token's key
    const float* kr = qkv + (size_t)b * QKV_ROWS + NUM_HEADS * HEAD_DIM + kvh * HEAD_DIM;
    float a0 = 0.f, a1 = 0.f;
    for (int d = 0; d < HEAD_DIM; ++d) { a0 += kr[d] * q_s[0][d]; a1 += kr[d] * q_s[1][d]; }
    probs[0][PAST_LEN] = a0; probs[1][PAST_LEN] = a1;
    lmax0 = fmaxf(lmax0, a0); lmax1 = fmaxf(lmax1, a1);
  }

  // ---- softmax: block max, exp, block sum (per head) ----
  float w0 = wred_max(lmax0), w1 = wred_max(lmax1);
  if (lane == 0) { red0[wid] = w0; red1[wid] = w1; }
  __syncthreads();
  if (wid == 0) {
    float t0 = (lane < 8) ? red0[lane] : -3.4e38f;
    float t1 = (lane < 8) ? red1[lane] : -3.4e38f;
    t0 = wred_max(t0); t1 = wred_max(t1);
    if (lane == 0) { gsh[0] = t0; gsh[1] = t1; }
  }
  __syncthreads();
  const float gm0 = gsh[0], gm1 = gsh[1];

  float s0 = 0.f, s1 = 0.f;
  for (int j = tid; j <= PAST_LEN; j += 256) {
    float p0 = __expf(probs[0][j] - gm0); probs[0][j] = p0; s0 += p0;
    float p1 = __expf(probs[1][j] - gm1); probs[1][j] = p1; s1 += p1;
  }
  s0 = wred_sum(s0); s1 = wred_sum(s1);
  if (lane == 0) { red0[wid] = s0; red1[wid] = s1; }
  __syncthreads();
  if (wid == 0) {
    float t0 = (lane < 8) ? red0[lane] : 0.f;
    float t1 = (lane < 8) ? red1[lane] : 0.f;
    t0 = wred_sum(t0); t1 = wred_sum(t1);
    if (lane == 0) { gsh[2] = 1.0f / t0; gsh[3] = 1.0f / t1; }
  }
  __syncthreads();
  const float inv0 = gsh[2], inv1 = gsh[3];

  // ---- P . V : stream V once for both heads ----
  const float* Vc = past_v + ((size_t)b * NUM_KV_HEADS + kvh) * (size_t)PAST_LEN * HEAD_DIM;
  const int d4    = (tid & 31) * 4;
  const int slice = tid >> 5;
  float4 a0 = {0.f, 0.f, 0.f, 0.f}, a1 = {0.f, 0.f, 0.f, 0.f};
  for (int j = slice * 512; j < slice * 512 + 512; ++j) {
    float4 vv = *(const float4*)(Vc + (size_t)j * HEAD_DIM + d4);
    float p0 = probs[0][j], p1 = probs[1][j];
    a0.x += p0 * vv.x; a0.y += p0 * vv.y; a0.z += p0 * vv.z; a0.w += p0 * vv.w;
    a1.x += p1 * vv.x; a1.y += p1 * vv.y; a1.z += p1 * vv.z; a1.w += p1 * vv.w;
  }
  pv[slice][0][d4 + 0] = a0.x; pv[slice][0][d4 + 1] = a0.y;
  pv[slice][0][d4 + 2] = a0.z; pv[slice][0][d4 + 3] = a0.w;
  pv[slice][1][d4 + 0] = a1.x; pv[slice][1][d4 + 1] = a1.y;
  pv[slice][1][d4 + 2] = a1.z; pv[slice][1][d4 + 3] = a1.w;
  __syncthreads();

  if (wid < 2) {   // wave 0 -> head h0, wave 1 -> head h0+1
    const int hh = wid;
    float4 tot = {0.f, 0.f, 0.f, 0.f};
#pragma unroll
    for (int s2 = 0; s2 < 8; ++s2) {
      tot.x += pv[s2][hh][d4 + 0]; tot.y += pv[s2][hh][d4 + 1];
      tot.z += pv[s2][hh][d4 + 2]; tot.w += pv[s2][hh][d4 + 3];
    }
    const float* vn = qkv + (size_t)b * QKV_ROWS +
                      (NUM_HEADS + NUM_KV_HEADS) * HEAD_DIM + kvh * HEAD_DIM;
    float pn = probs[hh][PAST_LEN];
    tot.x += pn * vn[d4 + 0]; tot.y += pn * vn[d4 + 1];
    tot.z += pn * vn[d4 + 2]; tot.w += pn * vn[d4 + 3];

    float invv = (hh == 0) ? inv0 : inv1;
    float* o = attn_out + (size_t)b * (NUM_HEADS * HEAD_DIM) + (h0 + hh) * HEAD_DIM + d4;
    o[0] = tot.x * invv; o[1] = tot.y * invv; o[2] = tot.z * invv; o[3] = tot.w * invv;
  }
}

// ---------------------------------------------------------------------------
extern "C" void kernel_launch(void* const* d_in, const int* in_sizes, int n_in,
                              void* d_out, int out_size, void* d_ws, size_t ws_size,
                              hipStream_t stream)
{
  const float* x      = (const float*)d_in[0];
  const float* past_k = (const float*)d_in[1];
  const float* past_v = (const float*)d_in[2];
  const float* Wq     = (const float*)d_in[3];
  const float* Wk     = (const float*)d_in[4];
  const float* Wv     = (const float*)d_in[5];
  const float* Wo     = (const float*)d_in[6];
  float* out = (float*)d_out;

  float* qkv  = (float*)d_ws;                        // 8*6144 f32
  float* attn = qkv + (size_t)BATCH * QKV_ROWS;      // 8*4096 f32
  float* part = attn + (size_t)BATCH * HIDDEN;       // split*8*6144 f32 (reused)

  // K-split factor: 8 if scratch allows (~1.9 MB), else 1 (deterministic:
  // depends only on ws_size, fixed for the session).
  const size_t base_f = (size_t)BATCH * QKV_ROWS + (size_t)BATCH * HIDDEN;
  int split = (ws_size >= (base_f + (size_t)8 * BATCH * QKV_ROWS) * sizeof(float)) ? 8 : 1;

  // ---- fused QKV projection: grid = (tiles/8, split) ----
  {
    dim3 grid((QKV_ROWS / 16 + 7) / 8, split);
    gemv16_wmma_split<<<grid, 256, 0, stream>>>(
        x, Wq, 4096, Wk, 1024, Wv, part, HIDDEN, QKV_ROWS, split);
    int n = BATCH * QKV_ROWS;
    reduce_split<<<(n + 255) / 256, 256, 0, stream>>>(part, qkv, QKV_ROWS, split, QKV_ROWS);
  }

  // ---- RoPE on q and new k ----
  int rope_threads = BATCH * (NUM_HEADS + NUM_KV_HEADS) * (HEAD_DIM / 2);
  rope_kernel<<<(rope_threads + 255) / 256, 256, 0, stream>>>(qkv);

  // ---- attention over 4096 cached + 1 new token ----
  dim3 ag(BATCH, NUM_KV_HEADS, 2);
  attn_kernel<<<ag, 256, 0, stream>>>(qkv, past_k, past_v, attn);

  // ---- output projection -> d_out [8][4096] ----
  {
    dim3 grid((HIDDEN / 16 + 7) / 8, split);
    gemv16_wmma_split<<<grid, 256, 0, stream>>>(
        attn, Wo, HIDDEN, Wo, 0, Wo, part, HIDDEN, HIDDEN, split);
    int n = BATCH * HIDDEN;
    reduce_split<<<(n + 255) / 256, 256, 0, stream>>>(part, out, HIDDEN, split, HIDDEN);
  }
}